// Mamba_46042049413756
// MI455X (gfx1250) — compile-verified
//
#include <hip/hip_runtime.h>
#include <hip/hip_bf16.h>
#include <math.h>

// ---------------- problem constants (from reference) ----------------
#define D_MODEL  2048
#define D_INNER  4096
#define D_XB     512
#define D_STATE  16
#define D_CONV   4
#define DT_RANK  128
#define BATCH    2
#define SEQLEN   1024
#define NXB      (D_XB / D_STATE)      // 32
#define NC       (D_INNER / D_STATE)   // 256
#define REP      (NC / NXB)            // 8
#define DPG      (D_INNER / NC)        // 16
#define ROWS     (BATCH * SEQLEN)      // 2048
#define LDZX     (2 * D_XB + 2 * D_INNER + DT_RANK)  // 9344
#define XOFF     (D_INNER)                           // 4096
#define BOFF     (D_INNER + D_XB)                    // 4608
#define COFF     (D_INNER + 2 * D_XB)                // 5120
#define DTOFF    (2 * D_INNER + 2 * D_XB)            // 9216

typedef float v2f __attribute__((ext_vector_type(2)));
typedef float v8f __attribute__((ext_vector_type(8)));

// ---------------- fp32 WMMA GEMM:  C[M,N] = A[M,K] @ B[N,K]^T ----------------
// Requires: M % 64 == 0, N % 128 == 0, K % 32 == 0 (true for all three GEMMs).
#define BM 64
#define BN 128
#define BK 32

__global__ __launch_bounds__(256) void wmma_gemm_f32(
    const float* __restrict__ A, const float* __restrict__ B, float* __restrict__ C,
    int M, int N, int K, int lda, int ldb, int ldc)
{
    __shared__ float lA[BM][BK + 1];
    __shared__ float lB[BN][BK + 1];

    const int tid  = threadIdx.x;
    const int wave = tid >> 5;          // 0..7
    const int lane = tid & 31;
    const int wm   = wave >> 2;         // 0..1  wave row  (32 rows of M each)
    const int wn   = wave & 3;          // 0..3  wave col  (32 cols of N each)
    const int hi   = lane >> 4;         // 0..1
    const int ln   = lane & 15;         // m for A-frag, n for B-frag / C cols

    const int block_m = blockIdx.y * BM;
    const int block_n = blockIdx.x * BN;

    v8f acc[2][2];
#pragma unroll
    for (int i = 0; i < 2; ++i)
#pragma unroll
        for (int j = 0; j < 2; ++j) acc[i][j] = (v8f)0.0f;

    // global-load coordinates (whole tile fully covered by 256 threads)
    const int ar = tid >> 2;              // 0..63   A row
    const int ac = (tid & 3) * 8;         // 8 floats per thread
    const int br = tid >> 1;              // 0..127  B row
    const int bc = (tid & 1) * 16;        // 16 floats per thread

    for (int k0 = 0; k0 < K; k0 += BK) {
        const float* Ap = A + (size_t)(block_m + ar) * lda + k0 + ac;
        const float* Bp = B + (size_t)(block_n + br) * ldb + k0 + bc;
#pragma unroll
        for (int i = 0; i < 8; ++i)  lA[ar][ac + i] = Ap[i];
#pragma unroll
        for (int i = 0; i < 16; ++i) lB[br][bc + i] = Bp[i];
        __syncthreads();

#pragma unroll
        for (int kk = 0; kk < BK; kk += 4) {
            // A fragment (16x4 f32): lane(hi,m) holds K = kk + 2*hi + {0,1}
            v2f afr[2], bfr[2];
#pragma unroll
            for (int i = 0; i < 2; ++i) {
                const int m = wm * 32 + i * 16 + ln;
                afr[i].x = lA[m][kk + 2 * hi + 0];
                afr[i].y = lA[m][kk + 2 * hi + 1];
            }
            // B fragment (4x16 f32): lane(hi,n) holds K = kk + 2*hi + {0,1}
            // of B^T (i.e. W row n)
#pragma unroll
            for (int j = 0; j < 2; ++j) {
                const int n = wn * 32 + j * 16 + ln;
                bfr[j].x = lB[n][kk + 2 * hi + 0];
                bfr[j].y = lB[n][kk + 2 * hi + 1];
            }
#pragma unroll
            for (int i = 0; i < 2; ++i)
#pragma unroll
                for (int j = 0; j < 2; ++j)
                    acc[i][j] = __builtin_amdgcn_wmma_f32_16x16x4_f32(
                        false, afr[i], false, bfr[j], (short)0, acc[i][j], false, false);
        }
        __syncthreads();
    }

    // store: C lane(hi,n), VGPR r -> row (r + 8*hi)
#pragma unroll
    for (int i = 0; i < 2; ++i)
#pragma unroll
        for (int j = 0; j < 2; ++j) {
            const int m0 = block_m + wm * 32 + i * 16 + 8 * hi;
            const int n0 = block_n + wn * 32 + j * 16 + ln;
#pragma unroll
            for (int r = 0; r < 8; ++r)
                C[(size_t)(m0 + r) * ldc + n0] = acc[i][j][r];
        }
}

// ---------------- delta = softplus(dt_raw + 2*b_dt)  (in place) ----------------
__global__ __launch_bounds__(256) void delta_softplus_kernel(
    float* __restrict__ dt, const float* __restrict__ b_dt)
{
    const int i = blockIdx.x * 256 + threadIdx.x;
    const int c = i % D_INNER;
    float x = dt[i] + 2.0f * b_dt[c];
    dt[i] = (x > 20.0f) ? x : log1pf(expf(x));
}

// ---------------- repeat_kv + causal depthwise conv + SiLU ----------------
__global__ __launch_bounds__(256) void conv_silu_kernel(
    const float* __restrict__ zx, const float* __restrict__ conv_w,
    const float* __restrict__ conv_b, float* __restrict__ xconv)
{
    const int i   = blockIdx.x * 256 + threadIdx.x;
    const int c   = i % D_INNER;
    const int row = i / D_INNER;          // b*SEQLEN + t
    const int t   = row % SEQLEN;
    const int head = c >> 7;              // (c/16)/8 : source x head
    const int xc   = (head << 4) | (c & 15);
    float acc = conv_b[c];
#pragma unroll
    for (int j = 0; j < D_CONV; ++j) {
        const int tt = t + j - (D_CONV - 1);
        if (tt >= 0)
            acc += conv_w[c * D_CONV + j] *
                   zx[(size_t)(row + tt - t) * LDZX + XOFF + xc];
    }
    xconv[i] = acc / (1.0f + expf(-acc));   // silu
}

// ---------------- selective scan: one workgroup per (batch, group) ----------------
// thread = d*16 + n ; state kept in registers; n-reduction via 16-lane xor shuffle.
__global__ __launch_bounds__(256) void scan_kernel(
    const float* __restrict__ zx, const float* __restrict__ delta,
    const float* __restrict__ xconv, const float* __restrict__ A_log,
    const float* __restrict__ D_param, float* __restrict__ yact)
{
    const int bb = blockIdx.x / NC;       // batch
    const int g  = blockIdx.x % NC;       // group
    const int tid = threadIdx.x;
    const int d = tid >> 4;               // 0..15 channel within group
    const int n = tid & 15;               // 0..15 state index
    const int c = g * DPG + d;            // global channel
    const int bhead = g >> 3;             // B source head

    const float Aval = -expf(A_log[(size_t)c * D_STATE + n]);
    const float Dv   = D_param[c];

    float state = 0.0f;
    const int row0 = bb * SEQLEN;
    for (int t = 0; t < SEQLEN; ++t) {
        const size_t row  = (size_t)(row0 + t);
        const size_t rz   = row * LDZX;
        const float dlt = delta[row * D_INNER + c];
        const float u   = xconv[row * D_INNER + c];
        const float Bv  = zx[rz + BOFF + bhead * D_STATE + n];
        const float Cv  = zx[rz + COFF + g * D_STATE + n];

        state = expf(dlt * Aval) * state + (dlt * u) * Bv;

        float p = state * Cv;
        p += __shfl_xor(p, 1, 32);
        p += __shfl_xor(p, 2, 32);
        p += __shfl_xor(p, 4, 32);
        p += __shfl_xor(p, 8, 32);

        if (n == 0) {
            const float z  = zx[rz + c];                    // z slice (offset 0)
            const float sz = z / (1.0f + expf(-z));
            yact[row * D_INNER + c] = (p + u * Dv) * sz;
        }
    }
}

// ---------------- host launch ----------------
extern "C" void kernel_launch(void* const* d_in, const int* in_sizes, int n_in,
                              void* d_out, int out_size, void* d_ws, size_t ws_size,
                              hipStream_t stream)
{
    (void)in_sizes; (void)n_in; (void)out_size; (void)ws_size;
    const float* hs      = (const float*)d_in[0];
    const float* W_in    = (const float*)d_in[1];
    const float* W_dt    = (const float*)d_in[2];
    const float* b_dt    = (const float*)d_in[3];
    const float* conv_w  = (const float*)d_in[4];
    const float* conv_b  = (const float*)d_in[5];
    const float* A_log   = (const float*)d_in[6];
    const float* D_param = (const float*)d_in[7];
    const float* W_out   = (const float*)d_in[8];
    float* out = (float*)d_out;

    float* ws     = (float*)d_ws;
    float* zxbcdt = ws;                                        // ROWS * LDZX
    float* delta  = zxbcdt + (size_t)ROWS * LDZX;              // ROWS * D_INNER
    float* xconv  = delta  + (size_t)ROWS * D_INNER;           // ROWS * D_INNER
    float* yact   = xconv  + (size_t)ROWS * D_INNER;           // ROWS * D_INNER

    const dim3 blk(256);

    // 1) zxbcdt = hidden @ W_in^T        (2048 x 2048) x (9344 x 2048)^T
    wmma_gemm_f32<<<dim3(LDZX / BN, ROWS / BM), blk, 0, stream>>>(
        hs, W_in, zxbcdt, ROWS, LDZX, D_MODEL, D_MODEL, D_MODEL, LDZX);

    // 2) dt_raw = dtl @ W_dt^T           (2048 x 128) x (4096 x 128)^T
    wmma_gemm_f32<<<dim3(D_INNER / BN, ROWS / BM), blk, 0, stream>>>(
        zxbcdt + DTOFF, W_dt, delta, ROWS, D_INNER, DT_RANK, LDZX, DT_RANK, D_INNER);

    // 3) delta = softplus(dt_raw + 2*b_dt)   [reference adds bias twice]
    delta_softplus_kernel<<<(ROWS * D_INNER) / 256, blk, 0, stream>>>(delta, b_dt);

    // 4) xconv = silu(depthwise_causal_conv(repeat_kv(x)) + conv_b)
    conv_silu_kernel<<<(ROWS * D_INNER) / 256, blk, 0, stream>>>(
        zxbcdt, conv_w, conv_b, xconv);

    // 5) selective scan + D*u skip + silu(z) gating -> yact
    scan_kernel<<<BATCH * NC, blk, 0, stream>>>(
        zxbcdt, delta, xconv, A_log, D_param, yact);

    // 6) out = yact @ W_out^T            (2048 x 4096) x (2048 x 4096)^T
    wmma_gemm_f32<<<dim3(D_MODEL / BN, ROWS / BM), blk, 0, stream>>>(
        yact, W_out, out, ROWS, D_MODEL, D_INNER, D_INNER, D_INNER, D_MODEL);
}